// CannyEdge_83330955477492
// MI455X (gfx1250) — compile-verified
//
#include <hip/hip_runtime.h>
#include <hip/hip_bf16.h>
#include <math.h>

#define TILE_W 64
#define TILE_H 32
#define IMG_W  1024
#define OUT_H  1025
#define OUT_W  1025
#define S1W    68   // staged image tile width  (TILE_W + 4 halo)
#define S1H    36   // staged image tile height (TILE_H + 4 halo)
#define S2W    66   // gradient tile width  (TILE_W + 2 halo)
#define S2H    34   // gradient tile height (TILE_H + 2 halo)

// Compose the two reflect-pads:
//  blur grid index y reflects over [0,1024], then the blur->image
//  shift (-1) reflects over [0,1023].
__device__ __forceinline__ int mapB(int y) {
  if (y < 0) y = -y;
  if (y > 1024) y = 2048 - y;
  int i = y - 1;
  if (i < 0) i = -i;
  if (i > 1023) i = 2046 - i;
  return i;
}

__global__ __launch_bounds__(256) void canny_fused(const float* __restrict__ img,
                                                   float* __restrict__ out_image,
                                                   float* __restrict__ out_week,
                                                   float* __restrict__ out_sure) {
  __shared__ float simg[S1H * S1W];
  __shared__ float sgxy[S2H * S2W];
  __shared__ int   sbin[S2H * S2W];

  const int tid = threadIdx.x;
  const int tx0 = blockIdx.x * TILE_W;
  const int ty0 = blockIdx.y * TILE_H;
  const int n   = blockIdx.z;
  const float* imgn = img + (size_t)n * (IMG_W * IMG_W);

  // ---- Stage 1: async-stage reflect-mapped image tile into LDS ----
  for (int l = tid; l < S1H * S1W; l += 256) {
    int r = l / S1W, c = l - r * S1W;
    int iy = mapB(ty0 - 2 + r);
    int ix = mapB(tx0 - 2 + c);
    const float* src = imgn + (size_t)iy * IMG_W + ix;
    float* dst = &simg[l];
#if __has_builtin(__builtin_amdgcn_global_load_async_to_lds_b32)
    __builtin_amdgcn_global_load_async_to_lds_b32(
        (__attribute__((address_space(1))) int*)(uintptr_t)src,
        (__attribute__((address_space(3))) int*)(uintptr_t)dst,
        0, 0);
#else
    *dst = __builtin_nontemporal_load(src);
#endif
  }
#if __has_builtin(__builtin_amdgcn_s_wait_asynccnt)
  __builtin_amdgcn_s_wait_asynccnt(0);
#else
  asm volatile("s_wait_asynccnt 0" ::: "memory");
#endif
  __syncthreads();

  // ---- Stage 2: gradient magnitude + direction bin for tile + 1-halo ----
  for (int l = tid; l < S2H * S2W; l += 256) {
    int r = l / S2W, c = l - r * S2W;      // position (ty0-1+r, tx0-1+c)
    int y = ty0 - 1 + r, x = tx0 - 1 + c;
    float g = 0.0f;
    int   b = 0;
    if (y >= 0 && y < OUT_H && x >= 0 && x < OUT_W) {
      const float* s0 = &simg[(r    ) * S1W + c];
      const float* s1 = &simg[(r + 1) * S1W + c];
      const float* s2 = &simg[(r + 2) * S1W + c];
      float a00 = s0[0], a01 = s0[1], a02 = s0[2];
      float a10 = s1[0],              a12 = s1[2];
      float a20 = s2[0], a21 = s2[1], a22 = s2[2];
      float gx = (a02 - a00) + 2.0f * (a12 - a10) + (a22 - a20);
      float gy = (a20 + 2.0f * a21 + a22) - (a00 + 2.0f * a01 + a02);
      g = sqrtf(gx * gx + gy * gy);
      // Quantized gradient direction without atan2:
      //   theta = (atan2(gx,gy)*180/pi + 90) mod 180, bins of 45 deg at 22.5 offsets.
      //   tan(theta-90) = gx/gy, so classify on |gx| vs |gy| and the sign of gx*gy.
      const float T = 0.4142135623730951f;   // tan(22.5 deg)
      float ax = fabsf(gx), ay = fabsf(gy);
      if (ax < T * ay)      b = 2;           // theta ~ 90   : vertical neighbors
      else if (ay < T * ax) b = 0;           // theta ~ 0/180: horizontal neighbors
      else                  b = (gx * gy > 0.0f) ? 3 : 1;  // diagonals
    }
    sgxy[l] = g;
    sbin[l] = b;
  }
  __syncthreads();

  // ---- Stage 3: non-max suppression + hysteresis thresholds ----
  const size_t NPIX = (size_t)OUT_H * OUT_W;
  for (int l = tid; l < TILE_H * TILE_W; l += 256) {
    int ly = l >> 6, lx = l & (TILE_W - 1);
    int y = ty0 + ly, x = tx0 + lx;
    if (y >= OUT_H || x >= OUT_W) continue;
    int r = ly + 1, c = lx + 1;
    float cg = sgxy[r * S2W + c];
    int   cb = sbin[r * S2W + c];
    // neighbor direction per bin: 0:(0,-1) 1:(-1,+1) 2:(-1,0) 3:(-1,-1)
    int dy = (cb == 0) ? 0 : -1;
    int dx = (cb == 1) ? 1 : ((cb == 2) ? 0 : -1);
    int i1 = (r + dy) * S2W + (c + dx);
    int i2 = (r - dy) * S2W + (c - dx);
    float m1 = (sbin[i1] == cb) ? sgxy[i1] : 0.0f;
    float m2 = (sbin[i2] == cb) ? sgxy[i2] : 0.0f;
    float edge = (cg >= m1 && cg >= m2) ? cg : 0.0f;

    float sure  = (edge >= 100.0f) ? 255.0f : 0.0f;
    float week  = (edge >= 50.0f && edge < 100.0f) ? 255.0f : 0.0f;
    float image = (edge >= 50.0f) ? 255.5f : 0.0f;

    size_t o = (size_t)n * NPIX + (size_t)y * OUT_W + x;
    __builtin_nontemporal_store(image, out_image + o);
    __builtin_nontemporal_store(week,  out_week  + o);
    __builtin_nontemporal_store(sure,  out_sure  + o);
  }
}

extern "C" void kernel_launch(void* const* d_in, const int* in_sizes, int n_in,
                              void* d_out, int out_size, void* d_ws, size_t ws_size,
                              hipStream_t stream) {
  (void)in_sizes; (void)n_in; (void)out_size; (void)d_ws; (void)ws_size;
  const float* img = (const float*)d_in[0];
  float* out = (float*)d_out;
  const size_t NPIX = (size_t)16 * OUT_H * OUT_W;   // per output tensor
  float* out_image = out;
  float* out_week  = out + NPIX;
  float* out_sure  = out + 2 * NPIX;

  dim3 block(256, 1, 1);
  dim3 grid((OUT_W + TILE_W - 1) / TILE_W, (OUT_H + TILE_H - 1) / TILE_H, 16);
  canny_fused<<<grid, block, 0, stream>>>(img, out_image, out_week, out_sure);
}